// GNNRegressor_86655260164498
// MI455X (gfx1250) — compile-verified
//
#include <hip/hip_runtime.h>
#include <hip/hip_bf16.h>
#include <math.h>

typedef __attribute__((ext_vector_type(2))) float v2f;
typedef __attribute__((ext_vector_type(4))) float v4f;
typedef __attribute__((ext_vector_type(8))) float v8f;
typedef __attribute__((ext_vector_type(4))) unsigned int v4u;
typedef __attribute__((ext_vector_type(4))) int v4i;
typedef __attribute__((ext_vector_type(8))) int v8i;

#define N_NODES 50000
#define N_EDGES 640000
#define N_GRAPH 256
#define H       128
#define T_OUT   5
#define ALD     20    // padded leading dim for [64,16] edge-attr tile
#define MLD     132   // padded leading dim for [64,128] tiles

// ---------------------------------------------------------------------------
// f32 WMMA helpers (V_WMMA_F32_16X16X4_F32).
// A 16x4 per step: lane m=lane&15 holds K={k,k+1} (upper half-wave K+=2).
// B 4x16 symmetric. D: lane%16 = N, vgpr r -> M = r + 8*(lane>=16).
// B fragments are wave-invariant across row tiles -> load once into registers.
// A fragments are bulk-loaded (clause of ds_load_2addr_b64, single wait) and
// then consumed by an unbroken WMMA accumulation chain.
// ---------------------------------------------------------------------------
template <int KSTEPS>
__device__ __forceinline__ void load_b_frags(const float* __restrict__ B, int ldb,
                                             v2f* bf) {
  const int lane = threadIdx.x & 31;
  const int mn   = lane & 15;
  const int koff = (lane >> 4) << 1;
#pragma unroll
  for (int s = 0; s < KSTEPS; ++s) {
    const int k = s * 4 + koff;
    bf[s].x = B[k * ldb + mn];
    bf[s].y = B[(k + 1) * ldb + mn];
  }
}

template <int KSTEPS>
__device__ __forceinline__ v8f wmma_acc(const float* __restrict__ A, int lda,
                                        const v2f* bf, v8f acc) {
  const int lane = threadIdx.x & 31;
  const int mn   = lane & 15;
  const int koff = (lane >> 4) << 1;
  v2f af[KSTEPS];
#pragma unroll
  for (int s = 0; s < KSTEPS; ++s) {
    const int k = s * 4 + koff;
    af[s].x = A[mn * lda + k];
    af[s].y = A[mn * lda + k + 1];
  }
#pragma unroll
  for (int s = 0; s < KSTEPS; ++s)
    acc = __builtin_amdgcn_wmma_f32_16x16x4_f32(false, af[s], false, bf[s], (short)0,
                                                acc, false, false);
  return acc;
}

// ---------------------------------------------------------------------------
// Tensor Data Mover: DMA a 2-D f32 tensor (dim0 x dim1, row stride stride0
// elements) from global into LDS at byte offset lds_off. One call per wave.
// ---------------------------------------------------------------------------
__device__ __forceinline__ void tdm_load_2d_f32(const void* gptr, unsigned lds_off,
                                                int dim0, int dim1, int stride0) {
  const unsigned long long ga = (unsigned long long)gptr;
  v4u g0;
  g0[0] = 1u;                                            // count=1, user descriptor
  g0[1] = lds_off;                                       // lds_addr (bytes)
  g0[2] = (unsigned)(ga & 0xFFFFFFFFu);                  // global_addr[31:0]
  g0[3] = (unsigned)((ga >> 32) & 0x01FFFFFFu) | (2u << 30);  // addr[56:32] | type=2
  v8i g1;
  g1[0] = (2 << 16);                                     // data_size = 4 bytes
  g1[1] = (int)((unsigned)dim0 << 16);                   // tensor_dim0[15:0] @63:48
  g1[2] = (int)(((unsigned)dim0 >> 16) | ((unsigned)dim1 << 16)); // dim0_hi|dim1_lo
  g1[3] = (int)(((unsigned)dim1 >> 16) | ((unsigned)dim0 << 16)); // dim1_hi|tile_dim0
  g1[4] = (int)((unsigned)dim1 & 0xFFFFu);               // tile_dim1 (tile_dim2=0)
  g1[5] = stride0;                                       // tensor_dim0_stride[31:0]
  g1[6] = 0;
  g1[7] = 0;
  const v4i z4 = {0, 0, 0, 0};
#if __has_include(<hip/amd_detail/amd_gfx1250_TDM.h>)
  const v8i z8 = {0, 0, 0, 0, 0, 0, 0, 0};
  __builtin_amdgcn_tensor_load_to_lds(g0, g1, z4, z4, z8, 0);
#else
  __builtin_amdgcn_tensor_load_to_lds(g0, g1, z4, z4, 0);
#endif
}

// ------- h = x @ node_w + node_b; also seeds z = h  ([N,64] -> [N,128]) -------
__global__ void node_proj_kernel(const float* __restrict__ x, const float* __restrict__ w,
                                 const float* __restrict__ b, float* __restrict__ h,
                                 float* __restrict__ z) {
  __shared__ __align__(16) float xT[64 * 68];
  const int r0 = blockIdx.x * 64;
  const int tid = threadIdx.x;
  const bool blockFull = (r0 + 64 <= N_NODES);
  {
    const int row = tid >> 2;
    const int cb  = (tid & 3) * 16;
    const int grow = r0 + row;
    if (blockFull) {
      for (int i = 0; i < 16; i += 4)
        *(v4f*)(&xT[row * 68 + cb + i]) = *(const v4f*)(x + (size_t)grow * 64 + cb + i);
    } else {
      for (int i = 0; i < 16; i += 4) {
        v4f v = {0.f, 0.f, 0.f, 0.f};
        if (grow < N_NODES) v = *(const v4f*)(x + (size_t)grow * 64 + cb + i);
        *(v4f*)(&xT[row * 68 + cb + i]) = v;
      }
    }
  }
  const int wv = tid >> 5, lane = tid & 31;
  const int n = lane & 15, mb = (lane >> 4) << 3;
  v2f bf[16];
  load_b_frags<16>(w + wv * 16, H, bf);
  const float bias = b[wv * 16 + n];
  __syncthreads();
  for (int rt = 0; rt < 4; ++rt) {
    v8f acc;
#pragma unroll
    for (int i = 0; i < 8; ++i) acc[i] = bias;
    acc = wmma_acc<16>(&xT[rt * 16 * 68], 68, bf, acc);
    if (blockFull) {
#pragma unroll
      for (int r = 0; r < 8; ++r) {
        const size_t idx = (size_t)(r0 + rt * 16 + mb + r) * H + wv * 16 + n;
        h[idx] = acc[r];
        z[idx] = acc[r];
      }
    } else {
#pragma unroll
      for (int r = 0; r < 8; ++r) {
        const int grow = r0 + rt * 16 + mb + r;
        if (grow < N_NODES) {
          const size_t idx = (size_t)grow * H + wv * 16 + n;
          h[idx] = acc[r];
          z[idx] = acc[r];
        }
      }
    }
  }
}

// ------- fused: e = MLP(edge_attr); m = relu(h[src]+e); z[dst] += m -------
__global__ void edge_message_kernel(const float* __restrict__ ea, const int* __restrict__ eidx,
                                    const float* __restrict__ w1, const float* __restrict__ b1,
                                    const float* __restrict__ w2, const float* __restrict__ b2,
                                    const float* __restrict__ h, float* __restrict__ z) {
  __shared__ __align__(16) float aT[64 * ALD];
  __shared__ __align__(16) float mid[64 * MLD];
  __shared__ int sSrc[64], sDst[64];
  const int e0 = blockIdx.x * 64;
  const int tid = threadIdx.x;
  if (tid < 64) {
    sSrc[tid] = eidx[e0 + tid];
    sDst[tid] = eidx[N_EDGES + e0 + tid];
  }
  {
    const int edge = tid >> 2;
    const int cb   = (tid & 3) * 4;
    v4f v = *(const v4f*)(ea + (size_t)(e0 + edge) * 16 + cb);
    *(v4f*)(&aT[edge * ALD + cb]) = v;
  }
  const int wv = tid >> 5, lane = tid & 31;
  const int n = lane & 15, mb = (lane >> 4) << 3;
  // layer 1: relu(attr @ w1 + b1) -> mid (LDS)
  v2f bf1[4];
  load_b_frags<4>(w1 + wv * 16, H, bf1);
  const float b1v = b1[wv * 16 + n];
  __syncthreads();
  for (int rt = 0; rt < 4; ++rt) {
    v8f acc;
#pragma unroll
    for (int i = 0; i < 8; ++i) acc[i] = b1v;
    acc = wmma_acc<4>(&aT[rt * 16 * ALD], ALD, bf1, acc);
#pragma unroll
    for (int r = 0; r < 8; ++r)
      mid[(rt * 16 + mb + r) * MLD + wv * 16 + n] = fmaxf(acc[r], 0.f);
  }
  // layer 2: e = mid @ w2 + b2, then scatter relu(h[src]+e) into z[dst]
  v2f bf2[32];
  load_b_frags<32>(w2 + wv * 16, H, bf2);
  const float b2v = b2[wv * 16 + n];
  const int dim = wv * 16 + n;
  __syncthreads();
  for (int rt = 0; rt < 4; ++rt) {
    v8f acc;
#pragma unroll
    for (int i = 0; i < 8; ++i) acc[i] = b2v;
    acc = wmma_acc<32>(&mid[rt * 16 * MLD], MLD, bf2, acc);
#pragma unroll
    for (int r = 0; r < 8; ++r) {
      const int edge = rt * 16 + mb + r;
      const float mval = h[(size_t)sSrc[edge] * H + dim] + acc[r];
      if (mval > 0.f) unsafeAtomicAdd(&z[(size_t)sDst[edge] * H + dim], mval);
    }
  }
}

// ------- h = relu(BN(relu(z @ w1 + b1) @ w2 + b2)); optional z reseed.
//         Weights staged into LDS by the Tensor Data Mover. -------
__global__ void node_mlp_kernel(const float* __restrict__ zin,
                                const float* __restrict__ w1, const float* __restrict__ b1,
                                const float* __restrict__ w2, const float* __restrict__ b2,
                                const float* __restrict__ gamma, const float* __restrict__ beta,
                                float* __restrict__ hout, float* __restrict__ zout) {
  extern __shared__ __align__(16) float smem[];
  float* zT  = smem;                    // 64 * MLD
  float* mid = smem + 64 * MLD;         // 64 * MLD
  float* w1s = smem + 2 * 64 * MLD;     // H*H, byte offset 67584
  float* w2s = w1s + H * H;             // H*H, byte offset 133120
  const int r0 = blockIdx.x * 64;
  const int tid = threadIdx.x;
  const bool blockFull = (r0 + 64 <= N_NODES);
  // Wave 0 kicks off the Tensor Data Mover for both weight matrices.
  if (tid < 32) {
    tdm_load_2d_f32(w1, 2u * 64u * MLD * 4u, H, H, H);
    tdm_load_2d_f32(w2, 2u * 64u * MLD * 4u + (unsigned)(H * H * 4), H, H, H);
  }
  {
    const int row = tid >> 2;
    const int cb  = (tid & 3) * 32;
    const int grow = r0 + row;
    if (blockFull) {
      for (int i = 0; i < 32; i += 4)
        *(v4f*)(&zT[row * MLD + cb + i]) = *(const v4f*)(zin + (size_t)grow * H + cb + i);
    } else {
      for (int i = 0; i < 32; i += 4) {
        v4f v = {0.f, 0.f, 0.f, 0.f};
        if (grow < N_NODES) v = *(const v4f*)(zin + (size_t)grow * H + cb + i);
        *(v4f*)(&zT[row * MLD + cb + i]) = v;
      }
    }
  }
  if (tid < 32) __builtin_amdgcn_s_wait_tensorcnt(0);
  __syncthreads();
  const int wv = tid >> 5, lane = tid & 31;
  const int n = lane & 15, mb = (lane >> 4) << 3;
  v2f bf[32];
  load_b_frags<32>(w1s + wv * 16, H, bf);
  const float b1v = b1[wv * 16 + n];
  for (int rt = 0; rt < 4; ++rt) {
    v8f acc;
#pragma unroll
    for (int i = 0; i < 8; ++i) acc[i] = b1v;
    acc = wmma_acc<32>(&zT[rt * 16 * MLD], MLD, bf, acc);
#pragma unroll
    for (int r = 0; r < 8; ++r)
      mid[(rt * 16 + mb + r) * MLD + wv * 16 + n] = fmaxf(acc[r], 0.f);
  }
  __syncthreads();
  load_b_frags<32>(w2s + wv * 16, H, bf);
  const float b2v = b2[wv * 16 + n];
  const float gv = gamma[wv * 16 + n], bv = beta[wv * 16 + n];
  const float bscale = rsqrtf(1.0f + 1e-5f);
  for (int rt = 0; rt < 4; ++rt) {
    v8f acc;
#pragma unroll
    for (int i = 0; i < 8; ++i) acc[i] = b2v;
    acc = wmma_acc<32>(&mid[rt * 16 * MLD], MLD, bf, acc);
    if (blockFull) {
#pragma unroll
      for (int r = 0; r < 8; ++r) {
        const size_t idx = (size_t)(r0 + rt * 16 + mb + r) * H + wv * 16 + n;
        const float val = fmaxf(acc[r] * bscale * gv + bv, 0.f);
        hout[idx] = val;
        if (zout) zout[idx] = val;
      }
    } else {
#pragma unroll
      for (int r = 0; r < 8; ++r) {
        const int grow = r0 + rt * 16 + mb + r;
        if (grow < N_NODES) {
          const size_t idx = (size_t)grow * H + wv * 16 + n;
          const float val = fmaxf(acc[r] * bscale * gv + bv, 0.f);
          hout[idx] = val;
          if (zout) zout[idx] = val;
        }
      }
    }
  }
}

// ------- gate = relu(h @ gw1 + gb1) @ gw2 + gb2  ([N,1]) -------
__global__ void gate_kernel(const float* __restrict__ h, const float* __restrict__ w1,
                            const float* __restrict__ b1, const float* __restrict__ w2,
                            const float* __restrict__ b2, float* __restrict__ gate) {
  const int node = blockIdx.x * 8 + (threadIdx.x >> 5);
  const int lane = threadIdx.x & 31;
  if (node >= N_NODES) return;
  const float* hr = h + (size_t)node * H;
  float d1 = 0.f, d2 = 0.f;
  for (int k = 0; k < H; ++k) {
    const float hk = hr[k];
    d1 += hk * w1[k * 64 + lane];
    d2 += hk * w1[k * 64 + lane + 32];
  }
  const float t1 = fmaxf(d1 + b1[lane], 0.f);
  const float t2 = fmaxf(d2 + b1[lane + 32], 0.f);
  float p = t1 * w2[lane] + t2 * w2[lane + 32];
  for (int off = 16; off; off >>= 1) p += __shfl_xor(p, off, 32);
  if (lane == 0) gate[node] = p + b2[0];
}

// ------- segment bounds from sorted batch -------
__global__ void bounds_init_kernel(int* __restrict__ sG, int* __restrict__ eG) {
  if (threadIdx.x < N_GRAPH) { sG[threadIdx.x] = 0; eG[threadIdx.x] = 0; }
}
__global__ void bounds_kernel(const int* __restrict__ batch, int* __restrict__ sG,
                              int* __restrict__ eG) {
  const int i = blockIdx.x * blockDim.x + threadIdx.x;
  if (i >= N_NODES) return;
  const int b = batch[i];
  if (i == 0 || batch[i - 1] != b) sG[b] = i;
  if (i == N_NODES - 1 || batch[i + 1] != b) eG[b] = i + 1;
}

// ------- per-graph softmax pooling + head MLP -------
__global__ void pool_head_kernel(const float* __restrict__ gate, const float* __restrict__ h,
                                 const int* __restrict__ sG, const int* __restrict__ eG,
                                 const float* __restrict__ hw1, const float* __restrict__ hb1,
                                 const float* __restrict__ hw2, const float* __restrict__ hb2,
                                 float* __restrict__ out) {
  __shared__ float red[8];
  __shared__ float part[256];
  __shared__ float gvec[H];
  __shared__ float midv[H];
  __shared__ float sMax, sDen;
  const int g = blockIdx.x;
  const int tid = threadIdx.x, lane = tid & 31, wv = tid >> 5;
  const int s = sG[g], e = eG[g];
  // max
  float v = -INFINITY;
  for (int i = s + tid; i < e; i += 256) v = fmaxf(v, gate[i]);
  for (int off = 16; off; off >>= 1) v = fmaxf(v, __shfl_xor(v, off, 32));
  if (lane == 0) red[wv] = v;
  __syncthreads();
  if (tid == 0) { float m = red[0]; for (int i = 1; i < 8; ++i) m = fmaxf(m, red[i]); sMax = m; }
  __syncthreads();
  const float gmax = sMax;
  // denom
  float sum = 0.f;
  for (int i = s + tid; i < e; i += 256) sum += expf(gate[i] - gmax);
  for (int off = 16; off; off >>= 1) sum += __shfl_xor(sum, off, 32);
  if (lane == 0) red[wv] = sum;
  __syncthreads();
  if (tid == 0) { float m = 0.f; for (int i = 0; i < 8; ++i) m += red[i]; sDen = m; }
  __syncthreads();
  const float den = sDen;
  // weighted feature sum (2 partial accumulators per dim)
  const int d = tid & 127, half = tid >> 7;
  float acc = 0.f;
  for (int i = s + half; i < e; i += 2) acc += expf(gate[i] - gmax) * h[(size_t)i * H + d];
  part[tid] = acc;
  __syncthreads();
  if (tid < H) gvec[tid] = (den > 0.f) ? (part[tid] + part[tid + 128]) / den : 0.f;
  __syncthreads();
  if (tid < H) {
    float a = hb1[tid];
    for (int k = 0; k < H; ++k) a += gvec[k] * hw1[k * H + tid];
    midv[tid] = fmaxf(a, 0.f);
  }
  __syncthreads();
  if (tid < T_OUT) {
    float a = hb2[tid];
    for (int k = 0; k < H; ++k) a += midv[k] * hw2[k * T_OUT + tid];
    out[g * T_OUT + tid] = a;
  }
}

extern "C" void kernel_launch(void* const* d_in, const int* in_sizes, int n_in,
                              void* d_out, int out_size, void* d_ws, size_t ws_size,
                              hipStream_t stream) {
  const float* x         = (const float*)d_in[0];
  const float* edge_attr = (const float*)d_in[1];
  const float* node_w    = (const float*)d_in[2];
  const float* node_b    = (const float*)d_in[3];
  const float* edge_w1   = (const float*)d_in[4];
  const float* edge_b1   = (const float*)d_in[5];
  const float* edge_w2   = (const float*)d_in[6];
  const float* edge_b2   = (const float*)d_in[7];
  const float* conv_w1   = (const float*)d_in[8];
  const float* conv_b1   = (const float*)d_in[9];
  const float* conv_w2   = (const float*)d_in[10];
  const float* conv_b2   = (const float*)d_in[11];
  const float* bn_gamma  = (const float*)d_in[12];
  const float* bn_beta   = (const float*)d_in[13];
  const float* gate_w1   = (const float*)d_in[14];
  const float* gate_b1   = (const float*)d_in[15];
  const float* gate_w2   = (const float*)d_in[16];
  const float* gate_b2   = (const float*)d_in[17];
  const float* head_w1   = (const float*)d_in[18];
  const float* head_b1   = (const float*)d_in[19];
  const float* head_w2   = (const float*)d_in[20];
  const float* head_b2   = (const float*)d_in[21];
  const int*   edge_index= (const int*)d_in[22];
  const int*   batch     = (const int*)d_in[23];
  float* out = (float*)d_out;

  float* h    = (float*)d_ws;                 // N*H
  float* z    = h + (size_t)N_NODES * H;      // N*H
  float* gate = z + (size_t)N_NODES * H;      // N
  int*   sG   = (int*)(gate + N_NODES);       // G
  int*   eG   = sG + N_GRAPH;                 // G

  const int nodeBlocks = (N_NODES + 63) / 64;          // 782
  const int edgeBlocks = N_EDGES / 64;                 // 10000 (exact)
  const size_t mlpLds  = (size_t)(2 * 64 * MLD + 2 * H * H) * sizeof(float);  // 198656 B

  node_proj_kernel<<<nodeBlocks, 256, 0, stream>>>(x, node_w, node_b, h, z);
  for (int l = 0; l < 4; ++l) {
    edge_message_kernel<<<edgeBlocks, 256, 0, stream>>>(edge_attr, edge_index,
        edge_w1, edge_b1, edge_w2, edge_b2, h, z);
    node_mlp_kernel<<<nodeBlocks, 256, mlpLds, stream>>>(z,
        conv_w1 + (size_t)l * H * H, conv_b1 + l * H,
        conv_w2 + (size_t)l * H * H, conv_b2 + l * H,
        bn_gamma + l * H, bn_beta + l * H, h, (l < 3) ? z : nullptr);
  }
  gate_kernel<<<(N_NODES + 7) / 8, 256, 0, stream>>>(h, gate_w1, gate_b1, gate_w2, gate_b2, gate);
  bounds_init_kernel<<<1, 256, 0, stream>>>(sG, eG);
  bounds_kernel<<<(N_NODES + 255) / 256, 256, 0, stream>>>(batch, sG, eG);
  pool_head_kernel<<<N_GRAPH, 256, 0, stream>>>(gate, h, sG, eG,
      head_w1, head_b1, head_w2, head_b2, out);
}